// SimpleGIN_87273735455432
// MI455X (gfx1250) — compile-verified
//
#include <hip/hip_runtime.h>
#include <math.h>

typedef __attribute__((ext_vector_type(2))) float v2f;
typedef __attribute__((ext_vector_type(4))) float v4f;
typedef __attribute__((ext_vector_type(8))) float v8f;

#define DHID 128
#define NPAD 132
#define LN_EPS 1e-5f

// ---------------------------------------------------------------------------
// agg = h   (float4 copy; only needed once, for layer 0's accumulator init)
// ---------------------------------------------------------------------------
__global__ void gin_copy4_kernel(const float4* __restrict__ src,
                                 float4* __restrict__ dst, int n4) {
  int i = blockIdx.x * blockDim.x + threadIdx.x;
  if (i < n4) dst[i] = src[i];
}

// ---------------------------------------------------------------------------
// Edge scatter: agg[dst[e]] += h[src[e]] + edge_attr[e]
// One wave per edge: 32 lanes x float4 = 128 floats; src/dst wave-uniform.
// edge_attr is a 320MB/layer one-shot stream -> non-temporal loads, so it
// doesn't evict the L2-resident h (51MB) / agg working set (192MB L2).
// ---------------------------------------------------------------------------
__global__ void gin_edge_scatter_kernel(const float* __restrict__ h,
                                        const float* __restrict__ ea,
                                        const int* __restrict__ srcIdx,
                                        const int* __restrict__ dstIdx,
                                        float* __restrict__ agg, int e) {
  long tid = (long)blockIdx.x * blockDim.x + threadIdx.x;
  long tot = (long)e * 32;
  if (tid >= tot) return;
  int eid = (int)(tid >> 5);
  int c = ((int)tid & 31) << 2;
  int s = srcIdx[eid];
  int d = dstIdx[eid];
  const v4f hv = *(const v4f*)(h + (size_t)s * DHID + c);
  const v4f ev =
      __builtin_nontemporal_load((const v4f*)(ea + (size_t)eid * DHID + c));
  float* o = agg + (size_t)d * DHID + c;
  atomicAdd(o + 0, hv.x + ev.x);
  atomicAdd(o + 1, hv.y + ev.y);
  atomicAdd(o + 2, hv.z + ev.z);
  atomicAdd(o + 3, hv.w + ev.w);
}

__device__ __forceinline__ float gelu_exact(float x) {
  return 0.5f * x * (1.0f + erff(x * 0.70710678118654752440f));
}

// ---------------------------------------------------------------------------
// Fused node MLP: h_out = GELU( LN2( GELU(LN1(agg@W1^T + b1)) @ W2^T + b2 ) + h_in )
// Block = 8 waves = 16 node rows; wave w computes cols [16w,16w+16) with
// V_WMMA_F32_16X16X4_F32 over K=128 (32 MACs per GEMM).
// If aggnext != nullptr, also stores h_out into the next layer's accumulator
// (in-place reuse of the agg buffer is safe: each block reads its rows into
// LDS first and blocks own disjoint rows).
// ---------------------------------------------------------------------------
__global__ __launch_bounds__(256) void gin_node_kernel(
    const float* __restrict__ agg, const float* __restrict__ hin,
    const float* __restrict__ W1, const float* __restrict__ b1,
    const float* __restrict__ g1, const float* __restrict__ be1,
    const float* __restrict__ W2, const float* __restrict__ b2,
    const float* __restrict__ g2, const float* __restrict__ be2,
    float* __restrict__ hout, float* __restrict__ aggnext, int nrows) {
  __shared__ float tileA[16][NPAD];  // A-tile for GEMM1, then C2 staging
  __shared__ float tileZ[16][NPAD];  // C1 staging / z (A-tile for GEMM2)

  const int t = threadIdx.x;
  const int wave = t >> 5;
  const int lane = t & 31;
  const int half = lane >> 4;
  const int lm = lane & 15;
  const int row0 = blockIdx.x * 16;
  const int n0 = wave * 16;

  // Load A tile: 16 rows x 128 cols, 16 threads/row, 8 floats/thread.
  {
    const int r = t >> 4;
    const int c0 = (t & 15) * 8;
    const int grow = row0 + r;
    if (grow < nrows) {
      const float4* s = (const float4*)(agg + (size_t)grow * DHID + c0);
      *(float4*)&tileA[r][c0] = s[0];
      *(float4*)&tileA[r][c0 + 4] = s[1];
    } else {
#pragma unroll
      for (int i = 0; i < 8; ++i) tileA[r][c0 + i] = 0.0f;
    }
  }
  __syncthreads();

  // ---- GEMM1: C = tileA @ W1^T ----
  v8f acc = {};
  {
    const float* wrow = W1 + (size_t)(n0 + lm) * DHID;  // W1[n][k], torch layout
#pragma unroll 4
    for (int k = 0; k < DHID; k += 4) {
      v2f a, b;
      a.x = tileA[lm][k + 2 * half + 0];
      a.y = tileA[lm][k + 2 * half + 1];
      b.x = wrow[k + 2 * half + 0];
      b.y = wrow[k + 2 * half + 1];
      acc = __builtin_amdgcn_wmma_f32_16x16x4_f32(false, a, false, b,
                                                  (short)0, acc, false, false);
    }
  }
  // Stage C + bias into tileZ (VGPR j -> rows j, j+8)
  {
    const float bv = b1[n0 + lm];
#pragma unroll
    for (int j = 0; j < 8; ++j) tileZ[j + 8 * half][n0 + lm] = acc[j] + bv;
  }
  __syncthreads();

  // ---- LayerNorm1 + GELU (in place on tileZ) ----
  {
    const int r = t >> 4;
    const int c0 = (t & 15) * 8;
    float v[8], s = 0.f, s2 = 0.f;
#pragma unroll
    for (int i = 0; i < 8; ++i) {
      v[i] = tileZ[r][c0 + i];
      s += v[i];
      s2 += v[i] * v[i];
    }
#pragma unroll
    for (int m = 1; m < 16; m <<= 1) {  // reduce within 16-lane half (wave32)
      s += __shfl_xor(s, m, 32);
      s2 += __shfl_xor(s2, m, 32);
    }
    const float mu = s * (1.0f / DHID);
    const float rs = rsqrtf(s2 * (1.0f / DHID) - mu * mu + LN_EPS);
#pragma unroll
    for (int i = 0; i < 8; ++i) {
      float x = (v[i] - mu) * rs * g1[c0 + i] + be1[c0 + i];
      tileZ[r][c0 + i] = gelu_exact(x);
    }
  }
  __syncthreads();

  // ---- GEMM2: C2 = tileZ @ W2^T ----
  v8f acc2 = {};
  {
    const float* wrow = W2 + (size_t)(n0 + lm) * DHID;
#pragma unroll 4
    for (int k = 0; k < DHID; k += 4) {
      v2f a, b;
      a.x = tileZ[lm][k + 2 * half + 0];
      a.y = tileZ[lm][k + 2 * half + 1];
      b.x = wrow[k + 2 * half + 0];
      b.y = wrow[k + 2 * half + 1];
      acc2 = __builtin_amdgcn_wmma_f32_16x16x4_f32(false, a, false, b,
                                                   (short)0, acc2, false, false);
    }
  }
  {
    const float bv = b2[n0 + lm];
#pragma unroll
    for (int j = 0; j < 8; ++j) tileA[j + 8 * half][n0 + lm] = acc2[j] + bv;
  }
  __syncthreads();

  // ---- LayerNorm2 + residual + GELU -> global ----
  {
    const int r = t >> 4;
    const int c0 = (t & 15) * 8;
    float v[8], s = 0.f, s2 = 0.f;
#pragma unroll
    for (int i = 0; i < 8; ++i) {
      v[i] = tileA[r][c0 + i];
      s += v[i];
      s2 += v[i] * v[i];
    }
#pragma unroll
    for (int m = 1; m < 16; m <<= 1) {
      s += __shfl_xor(s, m, 32);
      s2 += __shfl_xor(s2, m, 32);
    }
    const float mu = s * (1.0f / DHID);
    const float rs = rsqrtf(s2 * (1.0f / DHID) - mu * mu + LN_EPS);
    const int grow = row0 + r;
    if (grow < nrows) {
      const float* hr = hin + (size_t)grow * DHID + c0;
      float o[8];
#pragma unroll
      for (int i = 0; i < 8; ++i) {
        float x = (v[i] - mu) * rs * g2[c0 + i] + be2[c0 + i];
        o[i] = gelu_exact(x + hr[i]);
      }
      float4* dst = (float4*)(hout + (size_t)grow * DHID + c0);
      dst[0] = *(float4*)&o[0];
      dst[1] = *(float4*)&o[4];
      if (aggnext != nullptr) {  // pre-init next layer's accumulator with h_out
        float4* adst = (float4*)(aggnext + (size_t)grow * DHID + c0);
        adst[0] = *(float4*)&o[0];
        adst[1] = *(float4*)&o[4];
      }
    }
  }
}

// ---------------------------------------------------------------------------
extern "C" void kernel_launch(void* const* d_in, const int* in_sizes, int n_in,
                              void* d_out, int out_size, void* d_ws,
                              size_t ws_size, hipStream_t stream) {
  const float* h = (const float*)d_in[0];
  const float* ea = (const float*)d_in[1];
  // d_in[2] = batch (unused by reference output)
  const int* eidx = (const int*)d_in[3];
  const float* W1 = (const float*)d_in[4];
  const float* b1 = (const float*)d_in[5];
  const float* g1 = (const float*)d_in[6];
  const float* be1 = (const float*)d_in[7];
  const float* W2 = (const float*)d_in[8];
  const float* b2 = (const float*)d_in[9];
  const float* g2 = (const float*)d_in[10];
  const float* be2 = (const float*)d_in[11];
  float* out = (float*)d_out;

  const int n = in_sizes[0] / DHID;           // 100000 nodes
  const int e = in_sizes[1] / DHID;           // 625000 edges
  const int L = in_sizes[4] / (DHID * DHID);  // 3 layers

  float* agg = (float*)d_ws;
  float* hA = agg + (size_t)n * DHID;
  float* hB = hA + (size_t)n * DHID;
  float* pp[2] = {hA, hB};

  const int* srcIdx = eidx;      // edge_index[0]
  const int* dstIdx = eidx + e;  // edge_index[1]

  const int n4 = n * DHID / 4;
  const long eThreads = (long)e * 32;
  const int edgeBlocks = (int)((eThreads + 255) / 256);
  const int nodeBlocks = (n + 15) / 16;

  // Layer 0 accumulator init: agg = h_input
  gin_copy4_kernel<<<(n4 + 255) / 256, 256, 0, stream>>>(
      (const float4*)h, (float4*)agg, n4);

  const float* hin = h;
  for (int layer = 0; layer < L; ++layer) {
    const bool last = (layer == L - 1);
    float* hout = last ? out : pp[layer & 1];

    // 1) agg[dst] += h[src] + edge_attr   (agg already holds h from the
    //    layer-0 copy or the previous node kernel's dual store)
    gin_edge_scatter_kernel<<<edgeBlocks, 256, 0, stream>>>(
        hin, ea, srcIdx, dstIdx, agg, e);

    // 2) fused node MLP (WMMA GEMM -> LN -> GELU -> WMMA GEMM -> LN -> +h -> GELU)
    //    also dual-stores h_out into agg as next layer's accumulator init.
    gin_node_kernel<<<nodeBlocks, 256, 0, stream>>>(
        agg, hin,
        W1 + (size_t)layer * DHID * DHID, b1 + (size_t)layer * DHID,
        g1 + (size_t)layer * DHID, be1 + (size_t)layer * DHID,
        W2 + (size_t)layer * DHID * DHID, b2 + (size_t)layer * DHID,
        g2 + (size_t)layer * DHID, be2 + (size_t)layer * DHID,
        hout, last ? nullptr : agg, n);

    hin = hout;
  }
}